// AdaMLP_58076547776866
// MI455X (gfx1250) — compile-verified
//
#include <hip/hip_runtime.h>
#include <hip/hip_bf16.h>

// AdaMLP: out[b,k,:] = relu(slots[b,k,:] @ w1[e] + b1[e]) @ w2[e] + b2[e],
// e = indices[b,k].  B=16, K=64 -> 1024 slots, 64 experts, 256 -> 1024 -> 256.
//
// Expert-grouped batched GEMMs with V_WMMA_F32_16X16X4_F32 (f32-exact).
// Weights (128MB) read once -> HBM floor ~5.5us @ 23.3 TB/s.
// H is stored with per-expert 16-row padding so all tiles are full
// (branch-free epilogues / A loads).  Layer2 stages A via
// global_load_async_to_lds_b128 (ASYNCcnt) - CDNA5 async path.

#define DIM   256
#define RDIM  1024
#define NEXP  64
#define NSLOT 1024              // B*K
#define A1PITCH (DIM + 4)       // 260: 16B-aligned rows, conflict-free column reads
#define A2PITCH (RDIM + 4)      // 1028
#define HMAX   (NSLOT + NEXP * 15)   // 1984 padded rows max

typedef __attribute__((ext_vector_type(2))) float v2f;
typedef __attribute__((ext_vector_type(8))) float v8f;

// ---------------------------------------------------------------------------
// Kernel 1: per-expert grouping.  One block of 1024 threads.
// ws: perm[1024] @0, offs[65] @4096, offsP[65] @4608 (16-padded), H @8192.
// ---------------------------------------------------------------------------
__global__ __launch_bounds__(NSLOT)
void adamlp_group_kernel(const int* __restrict__ indices,
                         int* __restrict__ perm,
                         int* __restrict__ offs,
                         int* __restrict__ offsP) {
    __shared__ int cnt[NEXP];
    __shared__ int cur[NEXP];
    __shared__ int base[NEXP + 1];
    __shared__ int baseP[NEXP + 1];

    const int t = threadIdx.x;
    if (t < NEXP) { cnt[t] = 0; cur[t] = 0; }
    __syncthreads();

    const int e = indices[t] & (NEXP - 1);
    atomicAdd(&cnt[e], 1);
    __syncthreads();

    if (t == 0) {
        int s = 0, sp = 0;
        for (int i = 0; i < NEXP; ++i) {
            base[i]  = s;  s  += cnt[i];
            baseP[i] = sp; sp += (cnt[i] + 15) & ~15;
        }
        base[NEXP] = s; baseP[NEXP] = sp;
    }
    __syncthreads();

    if (t < NEXP + 1) { offs[t] = base[t]; offsP[t] = baseP[t]; }
    const int p = atomicAdd(&cur[e], 1);
    perm[base[e] + p] = t;
}

// ---------------------------------------------------------------------------
// Kernel 2: H = relu(X_grouped @ w1[e] + b1[e]).
// grid = (64, 8), block = 128 (4 waves).  Each wave owns TWO 16-col tiles
// (n0, n0+16): one A ds-load feeds two WMMAs.
// ---------------------------------------------------------------------------
__global__ __launch_bounds__(128)
void adamlp_layer1_kernel(const float* __restrict__ slots,
                          const float* __restrict__ w1,
                          const float* __restrict__ b1,
                          const int*  __restrict__ perm,
                          const int*  __restrict__ offs,
                          const int*  __restrict__ offsP,
                          float* __restrict__ H) {
    extern __shared__ float atile[];            // [16][A1PITCH]
    const int e    = blockIdx.x;
    const int wave = threadIdx.x >> 5;
    const int lane = threadIdx.x & 31;
    const int lo   = lane & 15;
    const int hi   = lane >> 4;
    const int n0   = blockIdx.y * 128 + wave * 32;

    const int beg   = offs[e];
    const int ne    = offs[e + 1] - beg;
    const int begP  = offsP[e];
    const int nePad = (ne + 15) & ~15;

    const float bias0 = b1[(size_t)e * RDIM + n0 + lo];
    const float bias1 = b1[(size_t)e * RDIM + n0 + 16 + lo];
    const float* wcol0 = w1 + (size_t)e * DIM * RDIM + (n0 + lo);
    const float* wcol1 = wcol0 + 16;

    for (int mt = 0; mt < nePad; mt += 16) {
        // ---- cooperative gather of 16 slot rows (zero-padded) into LDS ----
        for (int i = threadIdx.x; i < 16 * (DIM / 4); i += 128) {
            const int r  = i >> 6;              // DIM/4 = 64 chunks per row
            const int c4 = i & 63;
            float4 v = make_float4(0.f, 0.f, 0.f, 0.f);
            const int m = mt + r;
            if (m < ne)
                v = ((const float4*)(slots + (size_t)perm[beg + m] * DIM))[c4];
            *(float4*)&atile[r * A1PITCH + c4 * 4] = v;
        }
        __syncthreads();

        v8f acc0, acc1;
        #pragma unroll
        for (int i = 0; i < 8; ++i) { acc0[i] = bias0; acc1[i] = bias1; }

        #pragma unroll 4
        for (int k = 0; k < DIM; k += 4) {
            const int kk = k + 2 * hi;
            // A 16x4 f32: lanes0-15 row=lane K={k,k+1}; lanes16-31 K={k+2,k+3}
            v2f a = *(const v2f*)&atile[lo * A1PITCH + kk];
            v2f b0, b1v;
            b0.x  = wcol0[(size_t)kk * RDIM];
            b0.y  = wcol0[(size_t)(kk + 1) * RDIM];
            b1v.x = wcol1[(size_t)kk * RDIM];
            b1v.y = wcol1[(size_t)(kk + 1) * RDIM];
            acc0 = __builtin_amdgcn_wmma_f32_16x16x4_f32(
                false, a, false, b0, (short)0, acc0, false, false);
            acc1 = __builtin_amdgcn_wmma_f32_16x16x4_f32(
                false, a, false, b1v, (short)0, acc1, false, false);
        }

        // ---- ReLU + unconditional store (padded rows hold relu(bias)) ----
        float* hrow = H + (size_t)(begP + mt + 8 * hi) * RDIM + n0 + lo;
        #pragma unroll
        for (int i = 0; i < 8; ++i) {
            hrow[(size_t)i * RDIM]      = acc0[i] > 0.f ? acc0[i] : 0.f;
            hrow[(size_t)i * RDIM + 16] = acc1[i] > 0.f ? acc1[i] : 0.f;
        }
        __syncthreads();
    }
}

// ---------------------------------------------------------------------------
// Kernel 3: out = H @ w2[e] + b2[e], scattered back through perm.
// grid = (64, 4), block = 128 (4 waves), 16 cols per wave.
// A tile (16 x 1024 f32, ~66KB LDS) staged with global_load_async_to_lds_b128.
// ---------------------------------------------------------------------------
__global__ __launch_bounds__(128)
void adamlp_layer2_kernel(const float* __restrict__ H,
                          const float* __restrict__ w2,
                          const float* __restrict__ b2,
                          const int*  __restrict__ perm,
                          const int*  __restrict__ offs,
                          const int*  __restrict__ offsP,
                          float* __restrict__ out) {
    extern __shared__ float atile[];            // [16][A2PITCH]
    const int e    = blockIdx.x;
    const int wave = threadIdx.x >> 5;
    const int lane = threadIdx.x & 31;
    const int lo   = lane & 15;
    const int hi   = lane >> 4;
    const int n0   = blockIdx.y * 64 + wave * 16;

    const int beg   = offs[e];
    const int ne    = offs[e + 1] - beg;
    const int begP  = offsP[e];
    const int nePad = (ne + 15) & ~15;

    const float bias  = b2[(size_t)e * DIM + n0 + lo];
    const float* wcol = w2 + (size_t)e * RDIM * DIM + (n0 + lo);
    const unsigned long long hbase = (unsigned long long)H;

    for (int mt = 0; mt < nePad; mt += 16) {
        // ---- async-stage full 16-row A tile from H into LDS ----
        // 16 * 1024 floats = 4096 float4 chunks; 128 threads -> 32 rounds.
        #pragma unroll 4
        for (int rr = 0; rr < 32; ++rr) {
            const int i  = rr * 128 + threadIdx.x;
            const int r  = i >> 8;              // RDIM/4 = 256 chunks per row
            const int c4 = i & 255;
            const unsigned int lds_off =
                (unsigned int)((r * A2PITCH + c4 * 4) * 4);
            const unsigned int goff =
                (unsigned int)(((begP + mt + r) * RDIM + c4 * 4) * 4);
            asm volatile(
                "global_load_async_to_lds_b128 %0, %1, %2\n"
                :
                : "v"(lds_off), "v"(goff), "s"(hbase)
                : "memory");
        }
        asm volatile("s_wait_asynccnt 0x0" ::: "memory");
        __syncthreads();

        v8f acc;
        #pragma unroll
        for (int i = 0; i < 8; ++i) acc[i] = bias;

        #pragma unroll 4
        for (int k = 0; k < RDIM; k += 4) {
            const int kk = k + 2 * hi;
            v2f a = *(const v2f*)&atile[lo * A2PITCH + kk];
            v2f b;
            b.x = wcol[(size_t)kk * DIM];
            b.y = wcol[(size_t)(kk + 1) * DIM];
            acc = __builtin_amdgcn_wmma_f32_16x16x4_f32(
                false, a, false, b, (short)0, acc, false, false);
        }

        // ---- scatter valid rows to out via perm ----
        #pragma unroll
        for (int i = 0; i < 8; ++i) {
            const int m = mt + i + 8 * hi;
            if (m < ne) {
                const int slot = perm[beg + m];
                out[(size_t)slot * DIM + n0 + lo] = acc[i];
            }
        }
        __syncthreads();
    }
}

// ---------------------------------------------------------------------------
// Host-side launcher
// ---------------------------------------------------------------------------
extern "C" void kernel_launch(void* const* d_in, const int* in_sizes, int n_in,
                              void* d_out, int out_size, void* d_ws, size_t ws_size,
                              hipStream_t stream) {
    const float* slots   = (const float*)d_in[0];   // (16,64,256)
    const float* w1      = (const float*)d_in[1];   // (64,256,1024)
    const float* b1      = (const float*)d_in[2];   // (64,1024)
    const float* w2      = (const float*)d_in[3];   // (64,1024,256)
    const float* b2      = (const float*)d_in[4];   // (64,256)
    const int*   indices = (const int*)d_in[5];     // (16,64)

    char* ws    = (char*)d_ws;
    int*  perm  = (int*)(ws);                        // 1024 ints
    int*  offs  = (int*)(ws + 4096);                 // 65 ints
    int*  offsP = (int*)(ws + 4608);                 // 65 ints
    float* H    = (float*)(ws + 8192);               // HMAX*1024 floats (~8.1MB)

    adamlp_group_kernel<<<1, NSLOT, 0, stream>>>(indices, perm, offs, offsP);

    dim3 g1(NEXP, RDIM / 128);
    adamlp_layer1_kernel<<<g1, 128, 16 * A1PITCH * sizeof(float), stream>>>(
        slots, w1, b1, perm, offs, offsP, H);

    dim3 g2(NEXP, DIM / 64);
    adamlp_layer2_kernel<<<g2, 128, 16 * A2PITCH * sizeof(float), stream>>>(
        H, w2, b2, perm, offs, offsP, (float*)d_out);
}